// AttentionManifold_22643067584703
// MI455X (gfx1250) — compile-verified
//
#include <hip/hip_runtime.h>
#include <hip/hip_bf16.h>
#include <math.h>

// Problem constants (from reference): B=32, P=128, IN=64, OUT=32
#define BB    32
#define PP    128
#define INDIM 64
#define ODIM  32
#define NMAT  (BB * PP)          // 4096 matrices per tensor
#define MSZ   (ODIM * ODIM)      // 1024 floats per 32x32 matrix

typedef __attribute__((ext_vector_type(2))) float v2f;
typedef __attribute__((ext_vector_type(8))) float v8f;

// ---------------------------------------------------------------------------
// 32x32 * 32x32 -> 32x32 GEMM in LDS. blockDim.x == 128 (4 waves, wave32).
// Wave w computes C tile (w>>1, w&1). Uses V_WMMA_F32_16X16X4_F32.
// A,B,C must be distinct LDS buffers. All 128 threads must call (barriers).
// ---------------------------------------------------------------------------
__device__ __forceinline__ void mm32_lds(float* __restrict__ C,
                                         const float* __restrict__ A,
                                         const float* __restrict__ B,
                                         int lane, int wave) {
  const int ti = (wave >> 1) * 16;
  const int tj = (wave & 1) * 16;
  const int r  = lane & 15;
  const int kh = (lane >> 4) * 2;   // K half-select per f32 A/B fragment layout
  v8f c = {};
#pragma unroll
  for (int k0 = 0; k0 < 32; k0 += 4) {
    v2f a, b;
    a.x = A[(ti + r) * 32 + k0 + kh + 0];
    a.y = A[(ti + r) * 32 + k0 + kh + 1];
    b.x = B[(k0 + kh + 0) * 32 + tj + r];
    b.y = B[(k0 + kh + 1) * 32 + tj + r];
    c = __builtin_amdgcn_wmma_f32_16x16x4_f32(false, a, false, b, (short)0, c,
                                              false, false);
  }
  __syncthreads();  // all waves done reading A/B before C is overwritten
#pragma unroll
  for (int v = 0; v < 8; ++v)
    C[(ti + (lane >> 4) * 8 + v) * 32 + tj + r] = c[v];
  __syncthreads();
}

// ---------------------------------------------------------------------------
// Kernel 1: bimap  Q/K/V[b,p] = W * S * W^T   (one block per (b,p))
// ---------------------------------------------------------------------------
__global__ __launch_bounds__(256) void bimap_kernel(
    const float* __restrict__ x, const float* __restrict__ Wq,
    const float* __restrict__ Wk, const float* __restrict__ Wv,
    float* __restrict__ Qo, float* __restrict__ Ko, float* __restrict__ Vo) {
  __shared__ float sS[INDIM * INDIM];   // 16 KB
  __shared__ float sW[ODIM * INDIM];    //  8 KB
  __shared__ float sT[ODIM * INDIM];    //  8 KB
  const int mid  = blockIdx.x;          // b*P + p
  const int tid  = threadIdx.x;
  const int lane = tid & 31;
  const int wave = tid >> 5;

  const float* Sg = x + (size_t)mid * INDIM * INDIM;
  for (int i = tid; i < INDIM * INDIM; i += 256) sS[i] = Sg[i];

  const float* Ws[3] = {Wq, Wk, Wv};
  float* Os[3] = {Qo, Ko, Vo};

  for (int wsel = 0; wsel < 3; ++wsel) {
    __syncthreads();
    for (int i = tid; i < ODIM * INDIM; i += 256) sW[i] = Ws[wsel][i];
    __syncthreads();

    // T = W(32x64) @ S(64x64): 8 tiles, one per wave
    {
      const int ti = (wave >> 2) * 16, tj = (wave & 3) * 16;
      const int r = lane & 15, kh = (lane >> 4) * 2;
      v8f c = {};
#pragma unroll
      for (int k0 = 0; k0 < INDIM; k0 += 4) {
        v2f a, b;
        a.x = sW[(ti + r) * INDIM + k0 + kh + 0];
        a.y = sW[(ti + r) * INDIM + k0 + kh + 1];
        b.x = sS[(k0 + kh + 0) * INDIM + tj + r];
        b.y = sS[(k0 + kh + 1) * INDIM + tj + r];
        c = __builtin_amdgcn_wmma_f32_16x16x4_f32(false, a, false, b, (short)0,
                                                  c, false, false);
      }
#pragma unroll
      for (int v = 0; v < 8; ++v)
        sT[(ti + (lane >> 4) * 8 + v) * INDIM + tj + r] = c[v];
    }
    __syncthreads();

    // O = T(32x64) @ W^T(64x32): 4 tiles, waves 0..3 (wave-uniform branch)
    if (wave < 4) {
      const int ti = (wave >> 1) * 16, tj = (wave & 1) * 16;
      const int r = lane & 15, kh = (lane >> 4) * 2;
      v8f c = {};
#pragma unroll
      for (int k0 = 0; k0 < INDIM; k0 += 4) {
        v2f a, b;
        a.x = sT[(ti + r) * INDIM + k0 + kh + 0];
        a.y = sT[(ti + r) * INDIM + k0 + kh + 1];
        b.x = sW[(tj + r) * INDIM + k0 + kh + 0];  // W^T[k][n] = W[n][k]
        b.y = sW[(tj + r) * INDIM + k0 + kh + 1];
        c = __builtin_amdgcn_wmma_f32_16x16x4_f32(false, a, false, b, (short)0,
                                                  c, false, false);
      }
      float* O = Os[wsel] + (size_t)mid * MSZ;
#pragma unroll
      for (int v = 0; v < 8; ++v)
        O[(ti + (lane >> 4) * 8 + v) * ODIM + tj + r] = c[v];
    }
  }
}

// ---------------------------------------------------------------------------
// Kernel 2: logm of SPD 32x32 via trace-scaling + coupled Newton-Schulz
// square roots (inverse-free) + Mercator series. One block (128 thr) / matrix.
// Also emits ||logm||_F^2 to sq[mid].
// ---------------------------------------------------------------------------
#define NSQRT  3
#define NSITER 7
#define NTERM  6

__global__ __launch_bounds__(128) void logm_kernel(const float* __restrict__ in,
                                                   float* __restrict__ out,
                                                   float* __restrict__ sq) {
  __shared__ float sY[MSZ], sZ[MSZ], sT[MSZ], sU[MSZ], sV[MSZ];  // 20 KB
  __shared__ float sRed[128];
  __shared__ float sc;
  const int mid = blockIdx.x;
  const int tid = threadIdx.x, lane = tid & 31, wave = tid >> 5;
  const float* A = in + (size_t)mid * MSZ;

  for (int i = tid; i < MSZ; i += 128) sY[i] = A[i];
  __syncthreads();
  if (tid == 0) {
    float t = 0.f;
    for (int i = 0; i < ODIM; ++i) t += sY[i * 33];
    sc = t;
  }
  __syncthreads();
  const float c    = sc;
  const float invc = 1.0f / c;
  for (int i = tid; i < MSZ; i += 128) sY[i] *= invc;  // spectrum in (0,1]
  __syncthreads();

  // inverse scaling-squaring: NSQRT coupled Newton-Schulz square roots
  for (int s = 0; s < NSQRT; ++s) {
    for (int i = tid; i < MSZ; i += 128)
      sZ[i] = ((i >> 5) == (i & 31)) ? 1.0f : 0.0f;
    __syncthreads();
    for (int it = 0; it < NSITER; ++it) {
      mm32_lds(sT, sZ, sY, lane, wave);              // T' = Z @ Y
      for (int i = tid; i < MSZ; i += 128) {
        float d = ((i >> 5) == (i & 31)) ? 3.0f : 0.0f;
        sT[i] = 0.5f * (d - sT[i]);                  // T = 0.5(3I - Z Y)
      }
      __syncthreads();
      mm32_lds(sU, sY, sT, lane, wave);              // Ynew = Y @ T
      mm32_lds(sV, sT, sZ, lane, wave);              // Znew = T @ Z
      for (int i = tid; i < MSZ; i += 128) { sY[i] = sU[i]; sZ[i] = sV[i]; }
      __syncthreads();
    }
  }

  // E = M^(1/2^k) - I ; log series  L = E - E^2/2 + E^3/3 - ...
  for (int i = tid; i < MSZ; i += 128) {
    float e = sY[i] - (((i >> 5) == (i & 31)) ? 1.0f : 0.0f);
    sT[i] = e;   // E
    sU[i] = e;   // L accumulator
    sV[i] = e;   // E^n
  }
  __syncthreads();
  for (int n = 2; n <= NTERM; ++n) {
    mm32_lds(sZ, sV, sT, lane, wave);                // E^n
    const float coef = ((n & 1) ? 1.0f : -1.0f) / (float)n;
    for (int i = tid; i < MSZ; i += 128) {
      float p = sZ[i];
      sV[i] = p;
      sU[i] += coef * p;
    }
    __syncthreads();
  }

  // logm(A) = 2^NSQRT * L + log(c) * I ; store + Frobenius^2 reduce
  const float scale = (float)(1 << NSQRT);
  const float lc    = logf(c);
  float* O = out + (size_t)mid * MSZ;
  float acc = 0.f;
  for (int i = tid; i < MSZ; i += 128) {
    float r = scale * sU[i] + (((i >> 5) == (i & 31)) ? lc : 0.0f);
    O[i] = r;
    acc += r * r;
  }
  sRed[tid] = acc;
  __syncthreads();
  for (int off = 64; off > 0; off >>= 1) {
    if (tid < off) sRed[tid] += sRed[tid + off];
    __syncthreads();
  }
  if (tid == 0) sq[mid] = sRed[0];
}

// ---------------------------------------------------------------------------
// Kernel 3: qk tile + energy + weights.  One wave per 16x16 tile of (Pq,Pk).
// weights[b,m,n] = 1/(1+log1p(max(qq_m + kk_n - 2*<LQ_m,LK_n>, 0)))
// ---------------------------------------------------------------------------
__global__ __launch_bounds__(32) void qk_weights_kernel(
    const float* __restrict__ LQ, const float* __restrict__ LK,
    const float* __restrict__ sq, float* __restrict__ Wgt) {
  const int b = blockIdx.z, tm = blockIdx.y * 16, tn = blockIdx.x * 16;
  const int lane = threadIdx.x;
  const float* Aq = LQ + (size_t)b * PP * MSZ;
  const float* Ak = LK + (size_t)b * PP * MSZ;
  const int r = lane & 15, kh = (lane >> 4) * 2;
  v8f c = {};
#pragma unroll 4
  for (int k0 = 0; k0 < MSZ; k0 += 4) {
    v2f a, bb;
    a.x  = Aq[(tm + r) * MSZ + k0 + kh + 0];
    a.y  = Aq[(tm + r) * MSZ + k0 + kh + 1];
    bb.x = Ak[(tn + r) * MSZ + k0 + kh + 0];   // B = LK^T
    bb.y = Ak[(tn + r) * MSZ + k0 + kh + 1];
    c = __builtin_amdgcn_wmma_f32_16x16x4_f32(false, a, false, bb, (short)0, c,
                                              false, false);
  }
  const float* qq = sq + (size_t)b * PP;              // Q norms
  const float* kk = sq + (size_t)NMAT + (size_t)b * PP;  // K norms
#pragma unroll
  for (int v = 0; v < 8; ++v) {
    const int m = tm + (lane >> 4) * 8 + v;
    const int n = tn + r;
    float e = fmaxf(qq[m] + kk[n] - 2.0f * c[v], 0.0f);
    Wgt[((size_t)b * PP + m) * PP + n] = 1.0f / (1.0f + log1pf(e));
  }
}

// ---------------------------------------------------------------------------
// Kernel 4: softmax over query axis (dim -2), write transposed:
// Pt[b,i,j] = exp(W[b,j,i]-max_j)/sum_j.   One block per (b,i), 128 threads.
// ---------------------------------------------------------------------------
__global__ __launch_bounds__(128) void softmax_q_kernel(
    const float* __restrict__ Wgt, float* __restrict__ Pt) {
  __shared__ float sV[128];
  const int b = blockIdx.x >> 7;
  const int i = blockIdx.x & 127;
  const int j = threadIdx.x;
  const float w = Wgt[((size_t)b * PP + j) * PP + i];
  sV[j] = w;
  __syncthreads();
  for (int off = 64; off > 0; off >>= 1) {
    if (j < off) sV[j] = fmaxf(sV[j], sV[j + off]);
    __syncthreads();
  }
  const float mx = sV[0];
  __syncthreads();
  const float e = __expf(w - mx);
  sV[j] = e;
  __syncthreads();
  for (int off = 64; off > 0; off >>= 1) {
    if (j < off) sV[j] += sV[j + off];
    __syncthreads();
  }
  Pt[((size_t)b * PP + i) * PP + j] = e / sV[0];
}

// ---------------------------------------------------------------------------
// Kernel 5: mean_log[b] = Pt[b](128x128) @ LVflat[b](128x1024). Wave per tile.
// ---------------------------------------------------------------------------
__global__ __launch_bounds__(32) void meanlog_kernel(
    const float* __restrict__ Pt, const float* __restrict__ LV,
    float* __restrict__ ML) {
  const int b = blockIdx.z, tm = blockIdx.y * 16, tn = blockIdx.x * 16;
  const int lane = threadIdx.x;
  const float* Ap = Pt + (size_t)b * PP * PP;
  const float* Bv = LV + (size_t)b * PP * MSZ;
  const int r = lane & 15, kh = (lane >> 4) * 2;
  v8f c = {};
#pragma unroll 4
  for (int k0 = 0; k0 < PP; k0 += 4) {
    v2f a, bb;
    a.x  = Ap[(tm + r) * PP + k0 + kh + 0];
    a.y  = Ap[(tm + r) * PP + k0 + kh + 1];
    bb.x = Bv[(k0 + kh + 0) * MSZ + tn + r];
    bb.y = Bv[(k0 + kh + 1) * MSZ + tn + r];
    c = __builtin_amdgcn_wmma_f32_16x16x4_f32(false, a, false, bb, (short)0, c,
                                              false, false);
  }
  float* O = ML + (size_t)b * PP * MSZ;
#pragma unroll
  for (int v = 0; v < 8; ++v)
    O[(size_t)(tm + (lane >> 4) * 8 + v) * MSZ + tn + r] = c[v];
}

// ---------------------------------------------------------------------------
// Kernel 6: expm via scaling-squaring + Horner Taylor. One block per matrix.
// ---------------------------------------------------------------------------
#define SEXP    5
#define NTAYLOR 8

__global__ __launch_bounds__(128) void expm_kernel(const float* __restrict__ in,
                                                   float* __restrict__ out) {
  __shared__ float sX[MSZ], sP[MSZ], sW2[MSZ];  // 12 KB
  const int mid = blockIdx.x;
  const int tid = threadIdx.x, lane = tid & 31, wave = tid >> 5;
  const float* A = in + (size_t)mid * MSZ;
  const float s2 = 1.0f / (float)(1 << SEXP);
  for (int i = tid; i < MSZ; i += 128) {
    sX[i] = A[i] * s2;
    sP[i] = ((i >> 5) == (i & 31)) ? 1.0f : 0.0f;  // P = I
  }
  __syncthreads();
  // Horner: for n = NTAYLOR..1 : P = I + (X @ P) / n
  for (int n = NTAYLOR; n >= 1; --n) {
    mm32_lds(sW2, sX, sP, lane, wave);
    const float inv_n = 1.0f / (float)n;
    for (int i = tid; i < MSZ; i += 128)
      sP[i] = (((i >> 5) == (i & 31)) ? 1.0f : 0.0f) + sW2[i] * inv_n;
    __syncthreads();
  }
  // square SEXP times
  for (int s = 0; s < SEXP; ++s) {
    mm32_lds(sW2, sP, sP, lane, wave);
    for (int i = tid; i < MSZ; i += 128) sP[i] = sW2[i];
    __syncthreads();
  }
  float* O = out + (size_t)mid * MSZ;
  for (int i = tid; i < MSZ; i += 128) O[i] = sP[i];
}

// ---------------------------------------------------------------------------
extern "C" void kernel_launch(void* const* d_in, const int* in_sizes, int n_in,
                              void* d_out, int out_size, void* d_ws,
                              size_t ws_size, hipStream_t stream) {
  (void)in_sizes; (void)n_in; (void)out_size; (void)ws_size;
  const float* x  = (const float*)d_in[0];
  const float* Wq = (const float*)d_in[1];
  const float* Wk = (const float*)d_in[2];
  const float* Wv = (const float*)d_in[3];
  float* out = (float*)d_out;

  // Workspace layout (floats)
  float* ws   = (float*)d_ws;
  float* QKV  = ws;                                 // 3*4096*1024
  float* LQKV = QKV + (size_t)3 * NMAT * MSZ;       // 3*4096*1024
  float* SQ   = LQKV + (size_t)3 * NMAT * MSZ;      // 12288 (qq|kk|vv-unused)
  float* WG   = SQ + (size_t)3 * NMAT;              // 32*128*128
  float* PT   = WG + (size_t)BB * PP * PP;          // 32*128*128
  float* ML   = QKV;                                // reuse QKV (dead after logm)

  float* Qo = QKV;
  float* Ko = QKV + (size_t)NMAT * MSZ;
  float* Vo = QKV + (size_t)2 * NMAT * MSZ;
  float* LQ = LQKV;
  float* LK = LQKV + (size_t)NMAT * MSZ;
  float* LV = LQKV + (size_t)2 * NMAT * MSZ;

  // 1) Q/K/V = W S W^T
  bimap_kernel<<<NMAT, 256, 0, stream>>>(x, Wq, Wk, Wv, Qo, Ko, Vo);
  // 2) matrix logs (+ Frobenius norms)
  logm_kernel<<<3 * NMAT, 128, 0, stream>>>(QKV, LQKV, SQ);
  // 3) pairwise log-Euclidean energies -> weights
  qk_weights_kernel<<<dim3(PP / 16, PP / 16, BB), 32, 0, stream>>>(LQ, LK, SQ, WG);
  // 4) softmax over query axis, transposed output
  softmax_q_kernel<<<BB * PP, 128, 0, stream>>>(WG, PT);
  // 5) mean_log = prob @ LV
  meanlog_kernel<<<dim3(MSZ / 16, PP / 16, BB), 32, 0, stream>>>(PT, LV, ML);
  // 6) expm -> output
  expm_kernel<<<NMAT, 128, 0, stream>>>(ML, out);
}